// Model_61718680043882
// MI455X (gfx1250) — compile-verified
//
#include <hip/hip_runtime.h>
#include <math.h>

// ---------------------------------------------------------------------------
// CDNA5 (gfx1250) implementation of the 2x GAT-stack + MLP + X@Y^T reference.
// All dense matmuls use V_WMMA_F32_16X16X4_F32 (full fp32 precision on the
// matrix pipe).  Irregular edge-softmax stages use global float atomics.
// ---------------------------------------------------------------------------

typedef __attribute__((ext_vector_type(2))) float v2f;
typedef __attribute__((ext_vector_type(8))) float v8f;

#define CDIV(a, b) (((a) + (b)-1) / (b))

__device__ __forceinline__ void atomic_max_f32(float* addr, float v) {
  // Signed-float max via int/uint atomics (valid for mixed-sign, -inf init).
  if (v >= 0.0f)
    atomicMax((int*)addr, __float_as_int(v));
  else
    atomicMin((unsigned int*)addr, __float_as_uint(v));
}

// ---------------------------------------------------------------------------
// GEMM: C[N,Co] = maybe_relu(A[N,K] @ B + bias)
//   bt == 0 : B is [K,Co] row-major
//   bt == 1 : B is [Co,K] row-major  (computes A @ B^T, used for x @ y.T)
// One wave computes one 16x16 tile with K/4 V_WMMA_F32_16X16X4_F32 steps.
// N, Co multiples of 16; K multiple of 4 (true for all shapes here).
// ---------------------------------------------------------------------------
__global__ __launch_bounds__(256) void gemm_wmma_f32(
    const float* __restrict__ A, const float* __restrict__ B,
    const float* __restrict__ bias, float* __restrict__ C,
    int N, int K, int Co, int do_relu, int bt) {
  const int wave = (int)((blockIdx.x * blockDim.x + threadIdx.x) >> 5);
  const int lane = threadIdx.x & 31;
  const int tilesN = Co >> 4;
  const int tiles  = (N >> 4) * tilesN;
  if (wave >= tiles) return;  // wave-uniform: EXEC stays all-ones for WMMA

  const int tm = (wave / tilesN) << 4;
  const int tn = (wave % tilesN) << 4;
  const int g  = lane >> 4;   // lane group: carries K pair 2g, 2g+1
  const int r  = lane & 15;   // A row within tile / B,C column within tile

  v8f acc = {0.f, 0.f, 0.f, 0.f, 0.f, 0.f, 0.f, 0.f};
  const float* arow = A + (size_t)(tm + r) * K + 2 * g;

  if (bt) {
    const float* brow = B + (size_t)(tn + r) * K + 2 * g;
    for (int k = 0; k < K; k += 4) {
      v2f a, b;
      a.x = arow[k];     a.y = arow[k + 1];
      b.x = brow[k];     b.y = brow[k + 1];
      acc = __builtin_amdgcn_wmma_f32_16x16x4_f32(false, a, false, b,
                                                  (short)0, acc, false, false);
    }
  } else {
    for (int k = 0; k < K; k += 4) {
      v2f a, b;
      a.x = arow[k];     a.y = arow[k + 1];
      const float* bp = B + (size_t)(k + 2 * g) * Co + tn + r;
      b.x = bp[0];       b.y = bp[Co];
      acc = __builtin_amdgcn_wmma_f32_16x16x4_f32(false, a, false, b,
                                                  (short)0, acc, false, false);
    }
  }

  const float bv = bias ? bias[tn + r] : 0.0f;
#pragma unroll
  for (int v = 0; v < 8; ++v) {
    float val = acc[v] + bv;                       // row = tm + v + 8g, col = tn + r
    if (do_relu) val = fmaxf(val, 0.0f);
    C[(size_t)(tm + v + 8 * g) * Co + tn + r] = val;
  }
}

// alpha_src[n,h] = <h[n,h,:], a_src[h,:]>, alpha_dst likewise.
__global__ void node_alpha(const float* __restrict__ h,
                           const float* __restrict__ a_src,
                           const float* __restrict__ a_dst,
                           float* __restrict__ out_src,
                           float* __restrict__ out_dst, int N, int H, int C) {
  int t = blockIdx.x * blockDim.x + threadIdx.x;
  if (t >= N * H) return;
  int n = t / H, hh = t % H;
  const float* hv = h + (size_t)n * H * C + (size_t)hh * C;
  const float* as = a_src + (size_t)hh * C;
  const float* ad = a_dst + (size_t)hh * C;
  float s1 = 0.f, s2 = 0.f;
  for (int c = 0; c < C; ++c) {
    float v = hv[c];
    s1 += v * as[c];
    s2 += v * ad[c];
  }
  out_src[t] = s1;
  out_dst[t] = s2;
}

__global__ void init_softmax(float* __restrict__ emax, float* __restrict__ denom, int n) {
  int t = blockIdx.x * blockDim.x + threadIdx.x;
  if (t >= n) return;
  emax[t]  = -INFINITY;
  denom[t] = 0.0f;
}

// e = leaky_relu(alpha_src[src] + alpha_dst[dst]); segment-max into emax[dst].
__global__ void edge_logits(const int* __restrict__ src, const int* __restrict__ dst,
                            const float* __restrict__ a_s, const float* __restrict__ a_d,
                            float* __restrict__ e_buf, float* __restrict__ emax,
                            int E, int H) {
  int t = blockIdx.x * blockDim.x + threadIdx.x;
  if (t >= E * H) return;
  int e = t / H, hh = t % H;
  int j = src[e], i = dst[e];
  float v = a_s[j * H + hh] + a_d[i * H + hh];
  v = (v > 0.0f) ? v : 0.2f * v;  // leaky_relu, slope 0.2
  e_buf[t] = v;
  atomic_max_f32(&emax[i * H + hh], v);
}

// ee = exp(e - emax[dst]); segment-sum into denom[dst]; e_buf <- ee.
__global__ void edge_exp(const int* __restrict__ dst, float* __restrict__ e_buf,
                         const float* __restrict__ emax, float* __restrict__ denom,
                         int E, int H) {
  int t = blockIdx.x * blockDim.x + threadIdx.x;
  if (t >= E * H) return;
  int e = t / H, hh = t % H;
  int i  = dst[e];
  float ee = expf(e_buf[t] - emax[i * H + hh]);
  e_buf[t] = ee;
  atomicAdd(&denom[i * H + hh], ee);
}

// out[n, :] = bias[:]  (accumulator seeded with bias so relu(sum+b) works in place)
__global__ void init_bias(float* __restrict__ out, const float* __restrict__ bias,
                          int N, int HC) {
  size_t t = (size_t)blockIdx.x * blockDim.x + threadIdx.x;
  if (t >= (size_t)N * HC) return;
  out[t] = bias[t % HC];
}

// One wave per edge: out[dst] += h[src] * (ee / (denom[dst]+1e-16)) per head.
__global__ __launch_bounds__(256) void edge_aggregate(
    const int* __restrict__ src, const int* __restrict__ dst,
    const float* __restrict__ ee, const float* __restrict__ denom,
    const float* __restrict__ h, float* __restrict__ out, int E, int H, int C) {
  int wave = (int)((blockIdx.x * blockDim.x + threadIdx.x) >> 5);
  int lane = threadIdx.x & 31;
  if (wave >= E) return;
  int j = src[wave], i = dst[wave];
  int HC = H * C;
  for (int idx = lane; idx < HC; idx += 32) {
    int hh = idx / C;
    float w = ee[(size_t)wave * H + hh] / (denom[(size_t)i * H + hh] + 1e-16f);
    atomicAdd(&out[(size_t)i * HC + idx], h[(size_t)j * HC + idx] * w);
  }
}

__global__ void relu_ip(float* __restrict__ p, size_t n) {
  size_t t = (size_t)blockIdx.x * blockDim.x + threadIdx.x;
  if (t >= n) return;
  p[t] = fmaxf(p[t], 0.0f);
}

// ---------------------------------------------------------------------------
// Host-side orchestration
// ---------------------------------------------------------------------------
static inline void launch_gemm(const float* A, const float* B, const float* bias,
                               float* C, int N, int K, int Co, int relu, int bt,
                               hipStream_t s) {
  int tiles  = (N / 16) * (Co / 16);
  int blocks = CDIV(tiles * 32, 256);
  hipLaunchKernelGGL(gemm_wmma_f32, dim3(blocks), dim3(256), 0, s,
                     A, B, bias, C, N, K, Co, relu, bt);
}

static void run_gat(const float* x, int N, int Fin, const int* ei, int E,
                    const float* W, const float* a_s, const float* a_d,
                    const float* b, int H, int C, float* h_buf, float* e_buf,
                    float* al_s, float* al_d, float* emax, float* denom,
                    float* out, hipStream_t s) {
  const int HC = H * C;
  launch_gemm(x, W, nullptr, h_buf, N, Fin, HC, 0, 0, s);                 // h = x @ W
  hipLaunchKernelGGL(node_alpha, dim3(CDIV(N * H, 256)), dim3(256), 0, s,
                     h_buf, a_s, a_d, al_s, al_d, N, H, C);
  hipLaunchKernelGGL(init_softmax, dim3(CDIV(N * H, 256)), dim3(256), 0, s,
                     emax, denom, N * H);
  hipLaunchKernelGGL(edge_logits, dim3(CDIV(E * H, 256)), dim3(256), 0, s,
                     ei, ei + E, al_s, al_d, e_buf, emax, E, H);
  hipLaunchKernelGGL(edge_exp, dim3(CDIV(E * H, 256)), dim3(256), 0, s,
                     ei + E, e_buf, emax, denom, E, H);
  hipLaunchKernelGGL(init_bias, dim3((unsigned)CDIV((size_t)N * HC, 256)), dim3(256),
                     0, s, out, b, N, HC);
  hipLaunchKernelGGL(edge_aggregate, dim3(CDIV(E * 32, 256)), dim3(256), 0, s,
                     ei, ei + E, e_buf, denom, h_buf, out, E, H, C);
  hipLaunchKernelGGL(relu_ip, dim3((unsigned)CDIV((size_t)N * HC, 256)), dim3(256),
                     0, s, out, (size_t)N * HC);
}

extern "C" void kernel_launch(void* const* d_in, const int* in_sizes, int n_in,
                              void* d_out, int out_size, void* d_ws, size_t ws_size,
                              hipStream_t stream) {
  (void)in_sizes; (void)n_in; (void)out_size; (void)ws_size;
  const int Mn = 20000, Dn = 8000, FG = 64, EMM = 320000, EDD = 128000;

  const float* x_m   = (const float*)d_in[0];
  const float* x_d   = (const float*)d_in[1];
  const int*   mm_ei = (const int*)d_in[2];
  const int*   dd_ei = (const int*)d_in[3];
  const float* W_gx1 = (const float*)d_in[4];
  const float* as_gx1 = (const float*)d_in[5];
  const float* ad_gx1 = (const float*)d_in[6];
  const float* b_gx1  = (const float*)d_in[7];
  const float* W_gx2 = (const float*)d_in[8];
  const float* as_gx2 = (const float*)d_in[9];
  const float* ad_gx2 = (const float*)d_in[10];
  const float* b_gx2  = (const float*)d_in[11];
  const float* W_gy1 = (const float*)d_in[12];
  const float* as_gy1 = (const float*)d_in[13];
  const float* ad_gy1 = (const float*)d_in[14];
  const float* b_gy1  = (const float*)d_in[15];
  const float* W_gy2 = (const float*)d_in[16];
  const float* as_gy2 = (const float*)d_in[17];
  const float* ad_gy2 = (const float*)d_in[18];
  const float* b_gy2  = (const float*)d_in[19];
  const float* lx1_W = (const float*)d_in[20]; const float* lx1_b = (const float*)d_in[21];
  const float* lx2_W = (const float*)d_in[22]; const float* lx2_b = (const float*)d_in[23];
  const float* lx3_W = (const float*)d_in[24]; const float* lx3_b = (const float*)d_in[25];
  const float* ly1_W = (const float*)d_in[26]; const float* ly1_b = (const float*)d_in[27];
  const float* ly2_W = (const float*)d_in[28]; const float* ly2_b = (const float*)d_in[29];
  const float* ly3_W = (const float*)d_in[30]; const float* ly3_b = (const float*)d_in[31];

  // --- workspace layout (floats) -------------------------------------------
  float* w = (float*)d_ws;
  size_t off = 0;
  auto alloc = [&](size_t n) { float* p = w + off; off += n; return p; };

  float* h1x   = alloc((size_t)Mn * 256);
  float* X1    = alloc((size_t)Mn * 256);
  float* h2x   = alloc((size_t)Mn * 64);
  float* Xx    = alloc((size_t)Mn * 64);
  float* alsx  = alloc((size_t)Mn * 2);
  float* aldx  = alloc((size_t)Mn * 2);
  float* emaxx = alloc((size_t)Mn * 2);
  float* denx  = alloc((size_t)Mn * 2);
  float* e_x   = alloc((size_t)EMM * 2);
  float* mlp1x = alloc((size_t)Mn * 256);
  float* mlp2x = alloc((size_t)Mn * 128);
  float* featx = alloc((size_t)Mn * 64);

  float* h1y   = alloc((size_t)Dn * 256);
  float* Y1    = alloc((size_t)Dn * 256);
  float* h2y   = alloc((size_t)Dn * 64);
  float* Yy    = alloc((size_t)Dn * 64);
  float* alsy  = alloc((size_t)Dn * 2);
  float* aldy  = alloc((size_t)Dn * 2);
  float* emaxy = alloc((size_t)Dn * 2);
  float* deny  = alloc((size_t)Dn * 2);
  float* e_y   = alloc((size_t)EDD * 2);
  float* mlp1y = alloc((size_t)Dn * 256);
  float* mlp2y = alloc((size_t)Dn * 128);
  float* featy = alloc((size_t)Dn * 64);

  // --- X stack: gat_x1 (H=2,C=128) -> gat_x2 (H=1,C=64) -> MLP -------------
  run_gat(x_m, Mn, FG, mm_ei, EMM, W_gx1, as_gx1, ad_gx1, b_gx1, 2, 128,
          h1x, e_x, alsx, aldx, emaxx, denx, X1, stream);
  run_gat(X1, Mn, 256, mm_ei, EMM, W_gx2, as_gx2, ad_gx2, b_gx2, 1, 64,
          h2x, e_x, alsx, aldx, emaxx, denx, Xx, stream);
  launch_gemm(Xx,    lx1_W, lx1_b, mlp1x, Mn, 64,  256, 1, 0, stream);
  launch_gemm(mlp1x, lx2_W, lx2_b, mlp2x, Mn, 256, 128, 1, 0, stream);
  launch_gemm(mlp2x, lx3_W, lx3_b, featx, Mn, 128, 64,  1, 0, stream);

  // --- Y stack: gat_y1 -> gat_y2 -> MLP ------------------------------------
  run_gat(x_d, Dn, FG, dd_ei, EDD, W_gy1, as_gy1, ad_gy1, b_gy1, 2, 128,
          h1y, e_y, alsy, aldy, emaxy, deny, Y1, stream);
  run_gat(Y1, Dn, 256, dd_ei, EDD, W_gy2, as_gy2, ad_gy2, b_gy2, 1, 64,
          h2y, e_y, alsy, aldy, emaxy, deny, Yy, stream);
  launch_gemm(Yy,    ly1_W, ly1_b, mlp1y, Dn, 64,  256, 1, 0, stream);
  launch_gemm(mlp1y, ly2_W, ly2_b, mlp2y, Dn, 256, 128, 1, 0, stream);
  launch_gemm(mlp2y, ly3_W, ly3_b, featy, Dn, 128, 64,  1, 0, stream);

  // --- out = featx @ featy^T  [20000 x 8000] (store-bound; fp32 WMMA) ------
  launch_gemm(featx, featy, nullptr, (float*)d_out, Mn, 64, Dn, 0, 1, stream);
}